// GraphAttention_2439541424803
// MI455X (gfx1250) — compile-verified
//
#include <hip/hip_runtime.h>

#define N     8192
#define F_IN  256
#define F_OUT 128
#define ALPHA 0.2f

// k_hprime tiling
#define KB    64    // K-slab depth staged in LDS
#define PITCH 136   // padded LDS row pitch in words: (2*PITCH)%64 >= 16 -> no
                    // bank overlap between lane halves (khalf 0 vs 2)

typedef __attribute__((ext_vector_type(2))) float v2f;
typedef __attribute__((ext_vector_type(8))) float v8f;

// D = A(16x4, f32) * B(4x16, f32) + C(16x16, f32)  ->  v_wmma_f32_16x16x4_f32
static __device__ __forceinline__ v8f wmma_f32(v2f a, v2f b, v8f c) {
  return __builtin_amdgcn_wmma_f32_16x16x4_f32(
      /*neg_a=*/false, a, /*neg_b=*/false, b,
      /*c_mod=*/(short)0, c, /*reuse_a=*/false, /*reuse_b=*/false);
}

// Async global->LDS copy of 16 bytes (ASYNCcnt path, CDNA5 ISA).
// LDS byte address = low 32 bits of the flat address of a __shared__ object
// (LDS aperture rule).
static __device__ __forceinline__ void async_copy_b128(float* lds_ptr,
                                                       const float* gptr) {
  const unsigned ldst  = (unsigned)(size_t)lds_ptr;
  const unsigned long long gaddr = (unsigned long long)(size_t)gptr;
  asm volatile("global_load_async_to_lds_b128 %0, %1, off"
               :
               : "v"(ldst), "v"(gaddr)
               : "memory");
}
static __device__ __forceinline__ void wait_async0() {
  asm volatile("s_wait_asynccnt 0" ::: "memory");
}

// ---------------------------------------------------------------------------
// Kernel 1: Wh = h @ W   (8192x256 @ 256x128), f32 WMMA.
// Block = 256 threads = 8 waves; wave w computes the 16x16 tile at
// rows [blockIdx.x*16, +16) x cols [w*16, +16). K-loop: 64 WMMAs.
// ---------------------------------------------------------------------------
__global__ __launch_bounds__(256) void k_gemm_wh(const float* __restrict__ h,
                                                 const float* __restrict__ W,
                                                 float* __restrict__ Wh) {
  const int lane  = threadIdx.x & 31;
  const int wave  = threadIdx.x >> 5;
  const int rowB  = blockIdx.x * 16;
  const int colB  = wave * 16;
  const int m     = lane & 15;
  const int khalf = (lane >> 4) * 2;

  v8f acc = {};
  for (int k = 0; k < F_IN; k += 4) {
    v2f a, b;
    const float* hp = h + (size_t)(rowB + m) * F_IN + (k + khalf);
    a.x = hp[0];
    a.y = hp[1];
    const float* wp = W + (size_t)(k + khalf) * F_OUT + (colB + m);
    b.x = wp[0];
    b.y = wp[F_OUT];
    acc = wmma_f32(a, b, acc);
  }
  const int rhi = (lane >> 4) * 8;
#pragma unroll
  for (int r = 0; r < 8; ++r)
    Wh[(size_t)(rowB + r + rhi) * F_OUT + (colB + m)] = acc[r];
}

// ---------------------------------------------------------------------------
// Kernel 2: keep[i] = (sum_j adj[i][j] != 1.0).  Block per row, b128 loads.
// ---------------------------------------------------------------------------
__global__ __launch_bounds__(256) void k_keep(const float* __restrict__ adj,
                                              float* __restrict__ keepf) {
  __shared__ float red[256];
  const int i = blockIdx.x;
  const float4* row4 = (const float4*)(adj + (size_t)i * N);
  float s = 0.f;
  for (int j = threadIdx.x; j < N / 4; j += 256) {
    const float4 v = row4[j];
    s += (v.x + v.y) + (v.z + v.w);
  }
  red[threadIdx.x] = s;
  __syncthreads();
  for (int off = 128; off > 0; off >>= 1) {
    if (threadIdx.x < off) red[threadIdx.x] += red[threadIdx.x + off];
    __syncthreads();
  }
  if (threadIdx.x == 0) keepf[i] = (red[0] != 1.0f) ? 1.0f : 0.0f;
}

// ---------------------------------------------------------------------------
// Kernel 3: s_src = Wh @ a[:Fo], s_dst = Wh @ a[Fo:].  Thread per row.
// ---------------------------------------------------------------------------
__global__ __launch_bounds__(256) void k_scores(const float* __restrict__ Wh,
                                                const float* __restrict__ a,
                                                float* __restrict__ s_src,
                                                float* __restrict__ s_dst) {
  const int i = blockIdx.x * 256 + threadIdx.x;
  if (i >= N) return;
  const float* w = Wh + (size_t)i * F_OUT;
  float ss = 0.f, sd = 0.f;
  for (int c = 0; c < F_OUT; ++c) {
    const float v = w[c];
    ss += v * a[c];
    sd += v * a[F_OUT + c];
  }
  s_src[i] = ss;
  s_dst[i] = sd;
}

// ---------------------------------------------------------------------------
// Kernel 4: masked row softmax -> final_attn.  Block per row; e-row staged in
// 32KB LDS so adj is read once; all global streams use b128.
// ---------------------------------------------------------------------------
__global__ __launch_bounds__(256) void k_attn(const float* __restrict__ adj,
                                              const float* __restrict__ keepf,
                                              const float* __restrict__ s_src,
                                              const float* __restrict__ s_dst,
                                              float* __restrict__ attn) {
  __shared__ float ebuf[N];  // 32 KB
  __shared__ float red[256];
  const int   i      = blockIdx.x;
  const float keep_i = keepf[i];
  const float si     = s_src[i];

  const float4* arow4 = (const float4*)(adj + (size_t)i * N);
  const float4* keep4 = (const float4*)keepf;
  const float4* sdst4 = (const float4*)s_dst;
  float4*       e4    = (float4*)ebuf;

  auto elem = [&](float av, float kv, float sd) -> float {
    if ((av > 0.f) && (keep_i != 0.f) && (kv != 0.f)) {
      const float x = si + sd;
      return (x > 0.f) ? x : ALPHA * x;
    }
    return -1e30f;
  };

  // Pass 1: masked leaky-relu scores + running max
  float lmax = -1e30f;
  for (int j = threadIdx.x; j < N / 4; j += 256) {
    const float4 av = arow4[j];
    const float4 kv = keep4[j];
    const float4 sd = sdst4[j];
    float4 e;
    e.x = elem(av.x, kv.x, sd.x);
    e.y = elem(av.y, kv.y, sd.y);
    e.z = elem(av.z, kv.z, sd.z);
    e.w = elem(av.w, kv.w, sd.w);
    e4[j] = e;
    lmax = fmaxf(lmax, fmaxf(fmaxf(e.x, e.y), fmaxf(e.z, e.w)));
  }
  red[threadIdx.x] = lmax;
  __syncthreads();
  for (int off = 128; off > 0; off >>= 1) {
    if (threadIdx.x < off)
      red[threadIdx.x] = fmaxf(red[threadIdx.x], red[threadIdx.x + off]);
    __syncthreads();
  }
  float mrow = red[0];
  mrow = (mrow > -1e29f) ? mrow : 0.f;  // m = where(isfinite(m), m, 0)
  __syncthreads();

  // Pass 2: exp(e - m), running sum
  float lsum = 0.f;
  for (int j = threadIdx.x; j < N / 4; j += 256) {
    float4 e = e4[j];
    e.x = (e.x > -1e29f) ? __expf(e.x - mrow) : 0.f;
    e.y = (e.y > -1e29f) ? __expf(e.y - mrow) : 0.f;
    e.z = (e.z > -1e29f) ? __expf(e.z - mrow) : 0.f;
    e.w = (e.w > -1e29f) ? __expf(e.w - mrow) : 0.f;
    e4[j] = e;
    lsum += (e.x + e.y) + (e.z + e.w);
  }
  red[threadIdx.x] = lsum;
  __syncthreads();
  for (int off = 128; off > 0; off >>= 1) {
    if (threadIdx.x < off) red[threadIdx.x] += red[threadIdx.x + off];
    __syncthreads();
  }
  const float denom = red[0];
  const float inv   = (denom > 0.f) ? 1.f / fmaxf(denom, 1e-30f) : 0.f;

  // Pass 3: write normalized attention row (b128 stores)
  float4* out4 = (float4*)(attn + (size_t)i * N);
  for (int j = threadIdx.x; j < N / 4; j += 256) {
    float4 e = e4[j];
    e.x *= inv;
    e.y *= inv;
    e.z *= inv;
    e.w *= inv;
    out4[j] = e;
  }
  __syncthreads();
  if (threadIdx.x == 0 && keep_i == 0.f)
    attn[(size_t)i * N + i] = 1.0f;  // final_attn = attn + diag(~keep)
}

// ---------------------------------------------------------------------------
// Kernel 5: h_prime = keep ? (attn @ Wh) : Wh.
// Block = 128 threads (4 waves); wave w owns rows [blockRow + w*16, +16) and
// all 128 output cols (8 f32 accumulator tiles = 64 VGPRs).  Wh K-slabs are
// double-buffered in LDS via global_load_async_to_lds_b128 (ASYNCcnt), so Wh
// is read from L2 once per block and the 256MB attn stream exactly once.
// ---------------------------------------------------------------------------
__global__ __launch_bounds__(128) void k_hprime(const float* __restrict__ attn,
                                                const float* __restrict__ Wh,
                                                const float* __restrict__ keepf,
                                                float* __restrict__ out) {
  __shared__ float smem[2][KB * PITCH];  // 2 x 34KB

  const int tid   = threadIdx.x;
  const int lane  = tid & 31;
  const int wave  = tid >> 5;                 // 0..3
  const int rowB  = blockIdx.x * 64 + wave * 16;
  const int m     = lane & 15;
  const int khalf = (lane >> 4) * 2;

  // Cooperative async staging of one K-slab: KB rows x 128 cols of Wh.
  // 2048 float4 transfers / 128 threads = 16 per thread.
  auto stage_slab = [&](int k0, int buf) {
#pragma unroll
    for (int it = 0; it < 16; ++it) {
      const int q  = tid + it * 128;  // float4 index within slab
      const int kk = q >> 5;          // 32 float4 per 128-col row
      const int c4 = q & 31;
      async_copy_b128(&smem[buf][kk * PITCH + c4 * 4],
                      Wh + (size_t)(k0 + kk) * F_OUT + c4 * 4);
    }
  };

  v8f acc[8] = {};
  const float* arow = attn + (size_t)(rowB + m) * N;

  stage_slab(0, 0);
  const int NS = N / KB;  // 128 slabs
  for (int s = 0; s < NS; ++s) {
    wait_async0();      // own async copies landed
    __syncthreads();    // everyone's copies landed
    if (s + 1 < NS) stage_slab((s + 1) * KB, (s + 1) & 1);

    const float* bbuf = smem[s & 1];
    const int    kg   = s * KB;
    for (int kk = 0; kk < KB; kk += 4) {
      v2f a;
      a.x = arow[kg + kk + khalf];
      a.y = arow[kg + kk + khalf + 1];
#pragma unroll
      for (int t = 0; t < 8; ++t) {
        v2f b;
        const float* bp = bbuf + (kk + khalf) * PITCH + t * 16 + m;
        b.x = bp[0];
        b.y = bp[PITCH];
        acc[t] = wmma_f32(a, b, acc[t]);
      }
    }
    __syncthreads();    // all waves done reading buf before it is refilled
  }

  const int rhi = (lane >> 4) * 8;
#pragma unroll
  for (int r = 0; r < 8; ++r) {
    const int   row = rowB + r + rhi;
    const float kv  = keepf[row];
#pragma unroll
    for (int t = 0; t < 8; ++t) {
      const int col = t * 16 + m;
      const float v = (kv != 0.f) ? acc[t][r] : Wh[(size_t)row * F_OUT + col];
      out[(size_t)row * F_OUT + col] = v;
    }
  }
}

// ---------------------------------------------------------------------------
extern "C" void kernel_launch(void* const* d_in, const int* in_sizes, int n_in,
                              void* d_out, int out_size, void* d_ws,
                              size_t ws_size, hipStream_t stream) {
  const float* h   = (const float*)d_in[0];  // (N, F_IN)
  const float* adj = (const float*)d_in[1];  // (N, N)
  const float* W   = (const float*)d_in[2];  // (F_IN, F_OUT)
  const float* a   = (const float*)d_in[3];  // (2*F_OUT,)

  float* out_hprime = (float*)d_out;                   // (N, F_OUT)
  float* out_attn   = out_hprime + (size_t)N * F_OUT;  // (N, N)

  // Workspace: Wh (4MB) + s_src + s_dst + keep (~96KB)
  float* Wh    = (float*)d_ws;
  float* s_src = Wh + (size_t)N * F_OUT;
  float* s_dst = s_src + N;
  float* keepf = s_dst + N;

  k_gemm_wh<<<dim3(N / 16), dim3(256), 0, stream>>>(h, W, Wh);
  k_keep<<<dim3(N), dim3(256), 0, stream>>>(adj, keepf);
  k_scores<<<dim3(N / 256), dim3(256), 0, stream>>>(Wh, a, s_src, s_dst);
  k_attn<<<dim3(N), dim3(256), 0, stream>>>(adj, keepf, s_src, s_dst, out_attn);
  k_hprime<<<dim3(N / 64), dim3(128), 0, stream>>>(out_attn, Wh, keepf,
                                                   out_hprime);
}